// GravNetBlock_30666066493849
// MI455X (gfx1250) — compile-verified
//
#include <hip/hip_runtime.h>
#include <hip/hip_bf16.h>

// GravNetConv forward (GravNetBlock) for MI455X / gfx1250.
// B=64 graphs x L=1024 nodes, IN=128, OUT=128, S=4, FLR=64, K=16.
//
// Roofline: ~6 GFLOP total vs ~100 MB of streamed data -> memory/latency
// bound (~10us class at 23.3 TB/s). Wins: (a) never materialize the [B,L,L]
// distance matrix (saves ~268 MB HBM traffic), (b) fp32 WMMA
// (v_wmma_f32_16x16x4_f32) for all GEMMs to preserve reference numerics,
// (c) weights staged in LDS with bank-conflict-free strides (wave32 b32 read
// hits rows k / k+2 across the half-waves; stride st with 2*st == 32 mod 64
// puts the two halves in disjoint bank sets).

#define B_GR   64
#define L_GR   1024
#define N_TOT  (B_GR * L_GR)
#define IN_F   128
#define OUT_F  128
#define S_DIM  4
#define FLR_F  64
#define K_NN   16

// LDS row strides chosen so 2*stride == 32 (mod 64) -> conflict-free wave32 reads.
#define WS_ST  16     // padded W_s tile: 128 x 16, stride 16   (2*16 = 32)
#define WH_ST  80     // W_h tile:        128 x 64, stride 80   (2*80 = 160 = 32 mod 64)
#define WO_ST  144    // W_out chunk:      64 x 128, stride 144 (2*144 = 288 = 32 mod 64)

typedef __attribute__((ext_vector_type(2))) float v2f;
typedef __attribute__((ext_vector_type(8))) float v8f;

// 8-arg form: (neg_a, A, neg_b, B, c_mod, C, reuse_a, reuse_b)
#define WMMA_F32_16x16x4(a, b, c) \
  __builtin_amdgcn_wmma_f32_16x16x4_f32(false, (a), false, (b), (short)0, (c), false, false)

// ---------------------------------------------------------------------------
// Kernel 1: s = x @ W_s + b_s  (N x 4),  h = x @ W_h + b_h  (N x 64)
// One wave computes a 16-row slice. 5 WMMA accumulators: 1 (s, cols padded to
// 16 via an LDS tile -> branch-free inner loop) + 4 (h). 32 k-steps.
// blockDim = 256 (8 waves) -> 128 rows per block; grid = N/128 = 512.
// ---------------------------------------------------------------------------
__global__ void __launch_bounds__(256)
gravnet_sh_kernel(const float* __restrict__ x,
                  const float* __restrict__ Ws, const float* __restrict__ bs,
                  const float* __restrict__ Wh, const float* __restrict__ bh,
                  float* __restrict__ s_out, float* __restrict__ h_out) {
  __shared__ float sWs[IN_F * WS_ST];   // 8 KB, zero-padded to 16 cols
  __shared__ float sWh[IN_F * WH_ST];   // 40 KB

  const int tid  = threadIdx.x;
  const int lane = tid & 31;
  const int wave = tid >> 5;
  const int rowB = blockIdx.x * 128 + wave * 16;
  const int m    = lane & 15;          // A row within tile
  const int n    = lane & 15;          // B/C column within tile
  const int kb   = (lane >> 4) * 2;    // K sub-offset {0,2} per half-wave

  // Stage weights once per block (coalesced).
  for (int i = tid; i < IN_F * 16; i += 256) {          // padded W_s
    const int r = i >> 4, c = i & 15;
    sWs[r * WS_ST + c] = (c < S_DIM) ? Ws[r * S_DIM + c] : 0.0f;
  }
  for (int i = tid; i < IN_F * (FLR_F / 4); i += 256) { // W_h as float4
    const int r = i >> 4, c4 = i & 15;
    const float4 v = ((const float4*)(Wh + (size_t)r * FLR_F))[c4];
    *(float4*)(&sWh[r * WH_ST + c4 * 4]) = v;
  }
  __syncthreads();

  v8f accS = {0.f,0.f,0.f,0.f,0.f,0.f,0.f,0.f};
  v8f acc[4];
#pragma unroll
  for (int t = 0; t < 4; ++t) acc[t] = accS;

  const float* xrow = x + (size_t)(rowB + m) * IN_F;
  __builtin_prefetch(xrow, 0, 1);

  for (int kk = 0; kk < IN_F / 4; ++kk) {
    const int k = kk * 4 + kb;
    v2f a;
    a.x = xrow[k];
    a.y = xrow[k + 1];

    v2f bS;
    bS.x = sWs[k * WS_ST + n];
    bS.y = sWs[(k + 1) * WS_ST + n];
    accS = WMMA_F32_16x16x4(a, bS, accS);

#pragma unroll
    for (int t = 0; t < 4; ++t) {
      v2f bv;
      bv.x = sWh[k * WH_ST + t * 16 + n];
      bv.y = sWh[(k + 1) * WH_ST + t * 16 + n];
      acc[t] = WMMA_F32_16x16x4(a, bv, acc[t]);
    }
  }

  // C/D layout: VGPR r -> M = r (lanes 0-15) or r+8 (lanes 16-31), N = n.
  const int rOff = (lane >> 4) * 8;
#pragma unroll
  for (int r = 0; r < 8; ++r) {
    const int row = rowB + r + rOff;
    if (n < S_DIM) s_out[(size_t)row * S_DIM + n] = accS[r] + bs[n];
#pragma unroll
    for (int t = 0; t < 4; ++t)
      h_out[(size_t)row * FLR_F + t * 16 + n] = acc[t][r] + bh[t * 16 + n];
  }
}

// ---------------------------------------------------------------------------
// Kernel 2: per-node kNN (K=16, self included) in the 4-D learned space +
// distance-weighted mean/max aggregation of h.
// One thread = one node. Graph's s tile (1024x4 = 16 KB) staged in LDS.
// Top-16 kept in registers via unrolled insertion (strict < == stable
// tie-break by index, matching lax.top_k). grid = B * (L/256) = 256 blocks.
// h gathers are L2-resident (h = 16 MB << 192 MB L2).
// ---------------------------------------------------------------------------
__global__ void __launch_bounds__(256)
gravnet_knn_kernel(const float* __restrict__ s, const float* __restrict__ h,
                   float* __restrict__ agg_mean, float* __restrict__ agg_max) {
  __shared__ float4 sg[L_GR];           // 16 KB

  const int g     = blockIdx.x >> 2;    // graph id
  const int chunk = blockIdx.x & 3;     // 256-node chunk within graph
  const int tid   = threadIdx.x;

  const float4* sgl = (const float4*)(s + (size_t)g * L_GR * S_DIM);
  for (int i = tid; i < L_GR; i += 256) sg[i] = sgl[i];
  __syncthreads();

  const int l  = chunk * 256 + tid;     // node index within graph
  const float4 my = sg[l];

  float dist[K_NN];
  int   idx[K_NN];
#pragma unroll
  for (int j = 0; j < K_NN; ++j) { dist[j] = 3.0e38f; idx[j] = 0; }

  for (int c = 0; c < L_GR; ++c) {
    const float4 o = sg[c];
    const float dx = my.x - o.x, dy = my.y - o.y;
    const float dz = my.z - o.z, dw = my.w - o.w;
    const float d = dx * dx + dy * dy + dz * dz + dw * dw;
    if (d < dist[K_NN - 1]) {
      float dc = d; int ic = c;
#pragma unroll
      for (int j = 0; j < K_NN; ++j) {
        const bool sw = dc < dist[j];
        const float td = sw ? dist[j] : dc;
        const int   ti = sw ? idx[j]  : ic;
        dist[j] = sw ? dc : dist[j];
        idx[j]  = sw ? ic : idx[j];
        dc = td; ic = ti;
      }
    }
  }

  float w[K_NN];
#pragma unroll
  for (int j = 0; j < K_NN; ++j) w[j] = __expf(-10.0f * dist[j]);

  const float* hg = h + (size_t)g * L_GR * FLR_F;
  float* amean = agg_mean + ((size_t)g * L_GR + l) * FLR_F;
  float* amax  = agg_max  + ((size_t)g * L_GR + l) * FLR_F;

  for (int f = 0; f < FLR_F; f += 4) {
    float4 sum = {0.f, 0.f, 0.f, 0.f};
    float4 mx  = {-3.0e38f, -3.0e38f, -3.0e38f, -3.0e38f};
#pragma unroll
    for (int j = 0; j < K_NN; ++j) {
      const float4 hv = *(const float4*)(hg + (size_t)idx[j] * FLR_F + f);
      const float wj = w[j];
      const float m0 = hv.x * wj, m1 = hv.y * wj, m2 = hv.z * wj, m3 = hv.w * wj;
      sum.x += m0; sum.y += m1; sum.z += m2; sum.w += m3;
      mx.x = fmaxf(mx.x, m0); mx.y = fmaxf(mx.y, m1);
      mx.z = fmaxf(mx.z, m2); mx.w = fmaxf(mx.w, m3);
    }
    const float inv = 1.0f / (float)K_NN;
    *(float4*)(amean + f) = make_float4(sum.x * inv, sum.y * inv, sum.z * inv, sum.w * inv);
    *(float4*)(amax + f)  = mx;
  }
}

// ---------------------------------------------------------------------------
// Kernel 3: out = relu([x | agg_mean | agg_max] @ W_out + b_out)
// feats (K=256) never materialized. The 256-deep K loop runs as 4 uniform
// chunks (x-lo, x-hi, mean, max); each chunk's 64x128 W_out slice (36 KB) is
// staged in LDS. One wave: 16 rows x 128 cols = 8 f32 WMMA accumulators.
// blockDim = 256 (8 waves) -> 128 rows; grid = 512.
// ---------------------------------------------------------------------------
__global__ void __launch_bounds__(256)
gravnet_out_kernel(const float* __restrict__ x,
                   const float* __restrict__ agg_mean,
                   const float* __restrict__ agg_max,
                   const float* __restrict__ Wout, const float* __restrict__ bout,
                   float* __restrict__ out) {
  __shared__ float sW[64 * WO_ST];      // 36 KB per chunk

  const int tid  = threadIdx.x;
  const int lane = tid & 31;
  const int wave = tid >> 5;
  const int rowB = blockIdx.x * 128 + wave * 16;
  const int m    = lane & 15;
  const int n    = lane & 15;
  const int kb   = (lane >> 4) * 2;

  v8f acc[8];
#pragma unroll
  for (int t = 0; t < 8; ++t) acc[t] = (v8f){0.f,0.f,0.f,0.f,0.f,0.f,0.f,0.f};

  const float* xrow = x        + (size_t)(rowB + m) * IN_F;
  const float* mrow = agg_mean + (size_t)(rowB + m) * FLR_F;
  const float* xmax = agg_max  + (size_t)(rowB + m) * FLR_F;
  __builtin_prefetch(xrow, 0, 1);
  __builtin_prefetch(mrow, 0, 1);
  __builtin_prefetch(xmax, 0, 1);

  const float* srcs[4] = { xrow, xrow + 64, mrow, xmax };

  for (int c = 0; c < 4; ++c) {
    // Stage this chunk's 64x128 slice of W_out (rows c*64 .. c*64+63).
    for (int i = tid; i < 64 * (OUT_F / 4); i += 256) {  // 2048 float4
      const int r = i >> 5, c4 = i & 31;
      const float4 v = ((const float4*)(Wout + (size_t)(c * 64 + r) * OUT_F))[c4];
      *(float4*)(&sW[r * WO_ST + c4 * 4]) = v;
    }
    __syncthreads();

    const float* srcb = srcs[c];
    for (int kk = 0; kk < 16; ++kk) {
      const int kl = kk * 4 + kb;      // chunk-local k
      v2f a;
      a.x = srcb[kl];
      a.y = srcb[kl + 1];
#pragma unroll
      for (int t = 0; t < 8; ++t) {
        v2f bv;
        bv.x = sW[kl * WO_ST + t * 16 + n];
        bv.y = sW[(kl + 1) * WO_ST + t * 16 + n];
        acc[t] = WMMA_F32_16x16x4(a, bv, acc[t]);
      }
    }
    __syncthreads();
  }

  const int rOff = (lane >> 4) * 8;
#pragma unroll
  for (int r = 0; r < 8; ++r) {
    const int row = rowB + r + rOff;
#pragma unroll
    for (int t = 0; t < 8; ++t) {
      const float v = acc[t][r] + bout[t * 16 + n];
      out[(size_t)row * OUT_F + t * 16 + n] = fmaxf(v, 0.f);
    }
  }
}

// ---------------------------------------------------------------------------
extern "C" void kernel_launch(void* const* d_in, const int* in_sizes, int n_in,
                              void* d_out, int out_size, void* d_ws, size_t ws_size,
                              hipStream_t stream) {
  const float* x    = (const float*)d_in[0];   // [N, 128]
  const float* W_s  = (const float*)d_in[1];   // [128, 4]
  const float* b_s  = (const float*)d_in[2];   // [4]
  const float* W_h  = (const float*)d_in[3];   // [128, 64]
  const float* b_h  = (const float*)d_in[4];   // [64]
  const float* W_o  = (const float*)d_in[5];   // [256, 128]
  const float* b_o  = (const float*)d_in[6];   // [128]
  // d_in[7] = batch (int64) unused: graphs are contiguous, equal-size.

  float* out = (float*)d_out;                  // [N, 128]

  // Workspace layout (floats): s[N*4] | h[N*64] | mean[N*64] | max[N*64]
  float* s_ws   = (float*)d_ws;
  float* h_ws   = s_ws  + (size_t)N_TOT * S_DIM;
  float* mean_w = h_ws  + (size_t)N_TOT * FLR_F;
  float* max_w  = mean_w + (size_t)N_TOT * FLR_F;

  gravnet_sh_kernel<<<N_TOT / 128, 256, 0, stream>>>(x, W_s, b_s, W_h, b_h,
                                                     s_ws, h_ws);
  gravnet_knn_kernel<<<B_GR * (L_GR / 256), 256, 0, stream>>>(s_ws, h_ws,
                                                              mean_w, max_w);
  gravnet_out_kernel<<<N_TOT / 128, 256, 0, stream>>>(x, mean_w, max_w,
                                                      W_o, b_o, out);
}